// AllTnn_65962107732543
// MI455X (gfx1250) — compile-verified
//
#include <hip/hip_runtime.h>
#include <hip/hip_bf16.h>

typedef float v2f __attribute__((ext_vector_type(2)));
typedef float v8f __attribute__((ext_vector_type(8)));

// ---------------------------------------------------------------------------
// Fused LocallyConnected2D + ReLU + spatial LayerNorm.
// One block (256 threads) per batch sample; the oh*ow activation map lives in
// dynamic LDS so the LayerNorm reduction needs no extra global-memory pass.
// The per-position weights have zero reuse, so we prefetch the next output
// position's k*k filter block while the current FMA chain runs.
// ---------------------------------------------------------------------------
__global__ void lc_relu_ln_kernel(const float* __restrict__ in,
                                  const float* __restrict__ w,
                                  const float* __restrict__ bias,
                                  const float* __restrict__ gamma,
                                  const float* __restrict__ beta,
                                  float* __restrict__ out,
                                  int ih, int iw, int oh, int ow,
                                  int k, int stride)
{
    extern __shared__ float smem[];            // oh*ow activations
    __shared__ float s_sum[256];
    __shared__ float s_sq[256];

    const int b   = blockIdx.x;
    const int tid = threadIdx.x;
    const int N   = oh * ow;
    const int kk2 = k * k;
    const float* xin = in + (size_t)b * ih * iw;

    float lsum = 0.f, lsq = 0.f;
    for (int idx = tid; idx < N; idx += blockDim.x) {
        const int h    = idx / ow;
        const int wcol = idx - h * ow;
        const float* wp = w + (size_t)idx * kk2;
        const float* xp = xin + (h * stride) * iw + (wcol * stride);
        // prefetch the next filter block this thread will consume
        if (idx + (int)blockDim.x < N)
            __builtin_prefetch(w + (size_t)(idx + blockDim.x) * kk2, 0, 1);
        float acc = bias[idx];
        for (int r = 0; r < k; ++r) {
            const float* xr = xp + r * iw;
            for (int c = 0; c < k; ++c)
                acc = fmaf(xr[c], wp[r * k + c], acc);
        }
        acc = fmaxf(acc, 0.0f);                // ReLU
        smem[idx] = acc;
        lsum += acc;
        lsq  += acc * acc;
    }
    s_sum[tid] = lsum;
    s_sq[tid]  = lsq;
    __syncthreads();
    for (int off = 128; off > 0; off >>= 1) {
        if (tid < off) {
            s_sum[tid] += s_sum[tid + off];
            s_sq[tid]  += s_sq[tid + off];
        }
        __syncthreads();
    }
    const float invN = 1.0f / (float)N;
    const float mean = s_sum[0] * invN;
    const float var  = s_sq[0] * invN - mean * mean;
    const float rstd = rsqrtf(var + 1e-5f);

    float* op = out + (size_t)b * N;
    for (int idx = tid; idx < N; idx += blockDim.x)
        op[idx] = (smem[idx] - mean) * rstd * gamma[idx] + beta[idx];
}

// ---------------------------------------------------------------------------
// 2x2 maxpool, stride 2, VALID.
// ---------------------------------------------------------------------------
__global__ void maxpool2_kernel(const float* __restrict__ in,
                                float* __restrict__ out,
                                int B, int ih, int iw)
{
    const int oh = ih >> 1, ow = iw >> 1;
    const int total = B * oh * ow;
    for (int idx = blockIdx.x * blockDim.x + threadIdx.x; idx < total;
         idx += gridDim.x * blockDim.x) {
        int t = idx;
        const int wcol = t % ow; t /= ow;
        const int h    = t % oh; t /= oh;
        const int b    = t;
        const float* p = in + ((size_t)b * ih + 2 * h) * iw + 2 * wcol;
        out[idx] = fmaxf(fmaxf(p[0], p[1]), fmaxf(p[iw], p[iw + 1]));
    }
}

// ---------------------------------------------------------------------------
// FC: logits[128,1000] = A[128,64] * fcw[1000,64]^T + fcb, in fp32 via
// V_WMMA_F32_16X16X4_F32. One wave (32 threads) per 16x16 output tile,
// 16 K-steps of 4. N padded to 63 tiles.
// Out-of-range columns: clamp the row index to a legal address and cndmask
// the value to zero -- keeps the loads branch-free (no EXEC writes), so EXEC
// is provably all-1s at every WMMA as the ISA requires.
// A-frag:  lane l, vgpr v -> A[M = l%16, K = v + 2*(l/16)]
// B-frag:  lane l, vgpr v -> B[K = v + 2*(l/16), N = l%16]
// C/D:     lane l, vgpr r -> D[M = r + 8*(l/16), N = l%16]
// ---------------------------------------------------------------------------
__global__ void fc_wmma_kernel(const float* __restrict__ A,    // 128 x 64
                               const float* __restrict__ Wt,   // 1000 x 64
                               const float* __restrict__ bias, // 1000
                               float* __restrict__ logits)     // 128 x 1000
{
    const int lane = threadIdx.x;      // one wave per block (wave32)
    const int half = lane >> 4;
    const int lm   = lane & 15;
    const int m0 = blockIdx.x * 16;
    const int n0 = blockIdx.y * 16;
    const int n  = n0 + lm;
    const bool nok = (n < 1000);
    const int nc = nok ? n : 999;                     // always-legal address

    const float* arow = A  + (size_t)(m0 + lm) * 64 + 2 * half;
    const float* brow = Wt + (size_t)nc * 64 + 2 * half;

    v8f c = {};
    #pragma unroll
    for (int kk = 0; kk < 16; ++kk) {
        v2f a, bf;
        #pragma unroll
        for (int v = 0; v < 2; ++v) {
            const int kidx = kk * 4 + v;              // + 2*half folded in ptr
            a[v] = arow[kidx];
            const float t = brow[kidx];               // unconditional load
            bf[v] = nok ? t : 0.0f;                   // v_cndmask, no branch
        }
        // 8 args: (neg_a, A, neg_b, B, c_mod, C, reuse_a, reuse_b)
        c = __builtin_amdgcn_wmma_f32_16x16x4_f32(false, a, false, bf,
                                                  (short)0, c, false, false);
    }
    if (nok) {
        const float bn = bias[n];
        #pragma unroll
        for (int r = 0; r < 8; ++r) {
            const int m = m0 + r + 8 * half;
            logits[(size_t)m * 1000 + n] = c[r] + bn;
        }
    }
}

// ---------------------------------------------------------------------------
// Row softmax: one block (256 threads) per batch row of 1000 logits.
// Exponentials staged in LDS: one expf per element, one global read of logits.
// ---------------------------------------------------------------------------
__global__ void softmax_kernel(const float* __restrict__ logits,
                               float* __restrict__ out, int C)
{
    __shared__ float red[256];
    extern __shared__ float e[];                 // C floats
    const int row = blockIdx.x;
    const int tid = threadIdx.x;
    const float* lp = logits + (size_t)row * C;

    float m = -3.402823466e+38f;
    for (int i = tid; i < C; i += blockDim.x) {
        const float v = lp[i];
        e[i] = v;                                // stage raw logits
        m = fmaxf(m, v);
    }
    red[tid] = m; __syncthreads();
    for (int off = 128; off > 0; off >>= 1) {
        if (tid < off) red[tid] = fmaxf(red[tid], red[tid + off]);
        __syncthreads();
    }
    m = red[0];
    __syncthreads();

    float s = 0.f;
    for (int i = tid; i < C; i += blockDim.x) {
        const float ex = expf(e[i] - m);
        e[i] = ex;                               // overwrite with exp
        s += ex;
    }
    red[tid] = s; __syncthreads();
    for (int off = 128; off > 0; off >>= 1) {
        if (tid < off) red[tid] += red[tid + off];
        __syncthreads();
    }
    const float inv = 1.0f / red[0];

    float* op = out + (size_t)row * C;
    for (int i = tid; i < C; i += blockDim.x)
        op[i] = e[i] * inv;
}

// ---------------------------------------------------------------------------
extern "C" void kernel_launch(void* const* d_in, const int* in_sizes, int n_in,
                              void* d_out, int out_size, void* d_ws, size_t ws_size,
                              hipStream_t stream)
{
    const float* x = (const float*)d_in[0];
    const float* w1  = (const float*)d_in[1];
    const float* b1  = (const float*)d_in[2];
    const float* g1  = (const float*)d_in[3];
    const float* be1 = (const float*)d_in[4];
    const float* w2  = (const float*)d_in[5];
    const float* b2  = (const float*)d_in[6];
    const float* g2  = (const float*)d_in[7];
    const float* be2 = (const float*)d_in[8];
    const float* w3  = (const float*)d_in[9];
    const float* b3  = (const float*)d_in[10];
    const float* g3  = (const float*)d_in[11];
    const float* be3 = (const float*)d_in[12];
    const float* w4  = (const float*)d_in[13];
    const float* b4  = (const float*)d_in[14];
    const float* g4  = (const float*)d_in[15];
    const float* be4 = (const float*)d_in[16];
    const float* w5  = (const float*)d_in[17];
    const float* b5  = (const float*)d_in[18];
    const float* g5  = (const float*)d_in[19];
    const float* be5 = (const float*)d_in[20];
    const float* w6  = (const float*)d_in[21];
    const float* b6  = (const float*)d_in[22];
    const float* g6  = (const float*)d_in[23];
    const float* be6 = (const float*)d_in[24];
    const float* fcw = (const float*)d_in[25];
    const float* fcb = (const float*)d_in[26];
    float* out = (float*)d_out;

    const int B = 128;
    float* ws = (float*)d_ws;
    size_t off = 0;
    float* a1 = ws + off; off += (size_t)B * 109 * 109;
    float* p1 = ws + off; off += (size_t)B * 54 * 54;
    float* a2 = ws + off; off += (size_t)B * 50 * 50;
    float* a3 = ws + off; off += (size_t)B * 48 * 48;
    float* p3 = ws + off; off += (size_t)B * 24 * 24;
    float* a4 = ws + off; off += (size_t)B * 22 * 22;
    float* a5 = ws + off; off += (size_t)B * 20 * 20;
    float* p5 = ws + off; off += (size_t)B * 10 * 10;
    float* a6 = ws + off; off += (size_t)B * 8 * 8;
    float* lg = ws + off; off += (size_t)B * 1000;   // ~11.4 MB total

    const dim3 blk(256);

    // L1: 224 -> 109 (k=7, s=2)
    hipLaunchKernelGGL(lc_relu_ln_kernel, dim3(B), blk, 109 * 109 * sizeof(float),
                       stream, x, w1, b1, g1, be1, a1, 224, 224, 109, 109, 7, 2);
    // pool 109 -> 54
    hipLaunchKernelGGL(maxpool2_kernel, dim3((B * 54 * 54 + 255) / 256), blk, 0,
                       stream, a1, p1, B, 109, 109);
    // L2: 54 -> 50 (k=5)
    hipLaunchKernelGGL(lc_relu_ln_kernel, dim3(B), blk, 50 * 50 * sizeof(float),
                       stream, p1, w2, b2, g2, be2, a2, 54, 54, 50, 50, 5, 1);
    // L3: 50 -> 48 (k=3)
    hipLaunchKernelGGL(lc_relu_ln_kernel, dim3(B), blk, 48 * 48 * sizeof(float),
                       stream, a2, w3, b3, g3, be3, a3, 50, 50, 48, 48, 3, 1);
    // pool 48 -> 24
    hipLaunchKernelGGL(maxpool2_kernel, dim3((B * 24 * 24 + 255) / 256), blk, 0,
                       stream, a3, p3, B, 48, 48);
    // L4: 24 -> 22
    hipLaunchKernelGGL(lc_relu_ln_kernel, dim3(B), blk, 22 * 22 * sizeof(float),
                       stream, p3, w4, b4, g4, be4, a4, 24, 24, 22, 22, 3, 1);
    // L5: 22 -> 20
    hipLaunchKernelGGL(lc_relu_ln_kernel, dim3(B), blk, 20 * 20 * sizeof(float),
                       stream, a4, w5, b5, g5, be5, a5, 22, 22, 20, 20, 3, 1);
    // pool 20 -> 10
    hipLaunchKernelGGL(maxpool2_kernel, dim3((B * 10 * 10 + 255) / 256), blk, 0,
                       stream, a5, p5, B, 20, 20);
    // L6: 10 -> 8
    hipLaunchKernelGGL(lc_relu_ln_kernel, dim3(B), blk, 8 * 8 * sizeof(float),
                       stream, p5, w6, b6, g6, be6, a6, 10, 10, 8, 8, 3, 1);
    // FC via WMMA fp32: 8 x 63 tiles of 16x16, one wave each
    hipLaunchKernelGGL(fc_wmma_kernel, dim3(8, 63), dim3(32), 0, stream,
                       a6, fcw, fcb, lg);
    // softmax (dynamic LDS: 1000 floats for staged exponentials)
    hipLaunchKernelGGL(softmax_kernel, dim3(B), blk, 1000 * sizeof(float),
                       stream, lg, out, 1000);
}